// SparseAudioModel_9105330668181
// MI455X (gfx1250) — compile-verified
//
#include <hip/hip_runtime.h>

// SparseAudioModel: out[b,s] = sum_e x[b,e,s - 256*idx[b,e]]  (gather form of the
// reference scatter-add; s < S and t_e >= 0 make only the j >= 0 bound live).
//
// Memory-bound: 134 MB read / 2 MB write -> ~5.9 us at 23.3 TB/s HBM.
// The 64-event reduction runs through V_WMMA_F32_16X16X4_F32 with an all-ones
// A matrix (C[m,n] = sum_k B[k,n]): 4 events x 16 samples per WMMA, chained
// accumulation. Gathers are fully branchless: sign-mask clamp + bitwise zero,
// so loads form clauses instead of per-element exec-mask branches.

typedef __attribute__((ext_vector_type(2))) float v2f;
typedef __attribute__((ext_vector_type(8))) float v8f;

#define N_SAMPLES 32768
#define N_EVENTS  64
#define STEP      256
#define BATCH     16

#define WAVES_PER_BLOCK 8
#define WAVE_TILE       64   // samples per wave = 4 x (N=16) WMMA tiles
#define TILES_PER_BATCH  (N_SAMPLES / WAVE_TILE)              // 512
#define BLOCKS_PER_BATCH (TILES_PER_BATCH / WAVES_PER_BLOCK)  // 64
#define GRID (BATCH * BLOCKS_PER_BATCH)                       // 1024

__device__ __forceinline__ float masked_gather(const float* __restrict__ xb,
                                               int rowbase, int j) {
    const int mask = ~(j >> 31);                 // j<0 ? 0 : 0xFFFFFFFF
    const float v  = xb[rowbase + (j & mask)];   // clamped to row start: in-bounds
    return __int_as_float(__float_as_int(v) & mask);
}

__global__ __launch_bounds__(256)
void sparse_audio_gather_wmma(const float* __restrict__ x,
                              const int*   __restrict__ indices,
                              float*       __restrict__ out) {
    __shared__ int t_sh[N_EVENTS];

    const int tid = threadIdx.x;
    // 512 wave-tiles per batch, 8 waves per block -> whole block is one batch.
    const int b = blockIdx.x / BLOCKS_PER_BATCH;

    if (tid < N_EVENTS) {
        t_sh[tid] = indices[b * N_EVENTS + tid] * STEP;
    }
    __syncthreads();

    const int wave      = blockIdx.x * WAVES_PER_BLOCK + (tid >> 5);
    const int lane      = tid & 31;
    const int lh        = lane >> 4;     // lane half selects event within quad
    const int n_in_tile = lane & 15;     // N position inside a 16-wide tile
    const int s_base    = (wave % TILES_PER_BATCH) * WAVE_TILE;
    const int n0        = s_base + n_in_tile;

    const float* xb = x + (size_t)b * N_EVENTS * N_SAMPLES;

    v8f zero = {};
    v8f acc[4];
    acc[0] = zero; acc[1] = zero; acc[2] = zero; acc[3] = zero;

    v2f a_ones;
    a_ones.x = 1.0f;
    a_ones.y = 1.0f;

    #pragma unroll 8
    for (int g = 0; g < 16; ++g) {
        // events 4g..4g+3, split across lane halves and the two B VGPRs.
        // A == ones, so K-row placement is free; only N = lane&15 matters.
        const int eA = 4 * g + lh;       // 4g+0 (lanes 0-15) / 4g+1 (lanes 16-31)
        const int eB = 4 * g + 2 + lh;   // 4g+2 / 4g+3
        const int tA = t_sh[eA];
        const int tB = t_sh[eB];
        const int rowA = eA * N_SAMPLES; // 32-bit offsets: scalar-base + voffset
        const int rowB = eB * N_SAMPLES;

        #pragma unroll
        for (int t = 0; t < 4; ++t) {
            const int n = n0 + t * 16;
            v2f bm;
            bm.x = masked_gather(xb, rowA, n - tA);
            bm.y = masked_gather(xb, rowB, n - tB);
            // D = A(ones 16x4) * B(4x16 gathered) + C : sums 4 events per call
            acc[t] = __builtin_amdgcn_wmma_f32_16x16x4_f32(
                /*neg_a=*/false, a_ones,
                /*neg_b=*/false, bm,
                /*c_mod=*/(short)0, acc[t],
                /*reuse_a=*/false, /*reuse_b=*/false);
        }
    }

    // C/D layout: VGPR0, lanes 0-15 == row M=0, N=0..15 (all rows identical).
    if (lane < 16) {
        float* ob = out + (size_t)b * N_SAMPLES + s_base;
        #pragma unroll
        for (int t = 0; t < 4; ++t) {
            ob[t * 16 + lane] = acc[t][0];
        }
    }
}

extern "C" void kernel_launch(void* const* d_in, const int* in_sizes, int n_in,
                              void* d_out, int out_size, void* d_ws, size_t ws_size,
                              hipStream_t stream) {
    (void)in_sizes; (void)n_in; (void)out_size; (void)d_ws; (void)ws_size;
    const float* x   = (const float*)d_in[0];
    const int*   idx = (const int*)d_in[1];   // harness passes integers as int32
    float*       out = (float*)d_out;

    sparse_audio_gather_wmma<<<GRID, 256, 0, stream>>>(x, idx, out);
}